// PointConv2_53343493816566
// MI455X (gfx1250) — compile-verified
//
#include <hip/hip_runtime.h>
#include <hip/hip_bf16.h>

// ---------------------------------------------------------------------------
// PointConv: out[b,o,n] = sum_{c,k} x[b, c, idx[b,n,k]] * W[o,c,k] + bias[o]
// B=8, C_IN=C_OUT=32, N=65536, K=9.  fp32 end-to-end via V_WMMA_F32_16X16X4_F32.
// ---------------------------------------------------------------------------

typedef float v2f __attribute__((ext_vector_type(2)));
typedef float v8f __attribute__((ext_vector_type(8)));

constexpr int NPTS  = 65536;
constexpr int BATCH = 8;
constexpr int CIN   = 32;
constexpr int COUT  = 32;
constexpr int KNB   = 9;          // neighbors per point
constexpr int KFLAT = CIN * KNB;  // 288 (flattened as K = j*32 + c)
constexpr int WSTR  = 290;        // LDS row stride for W[o][K] (even -> b64 aligned)

// ---------------------------------------------------------------------------
// Pre-pass: transpose x[b][c][n] -> xt[b][n][c]  (each point's channel vector
// becomes one contiguous 128B line -> gathers hit single cachelines).
// ---------------------------------------------------------------------------
__global__ __launch_bounds__(256)
void transpose_xc(const float* __restrict__ x, float* __restrict__ xt) {
    __shared__ float tile[CIN][CIN + 1];
    const int b  = blockIdx.y;
    const int n0 = blockIdx.x * CIN;
    const int tx = threadIdx.x;          // 0..31
    const int ty = threadIdx.y;          // 0..7
    const float* xb = x + (size_t)b * CIN * NPTS;
#pragma unroll
    for (int i = 0; i < 4; ++i) {
        int c = ty * 4 + i;
        tile[c][tx] = xb[(size_t)c * NPTS + n0 + tx];   // coalesced over n
    }
    __syncthreads();
    float* xtb = xt + (size_t)b * NPTS * CIN;
#pragma unroll
    for (int i = 0; i < 4; ++i) {
        int nl = ty * 4 + i;
        xtb[(size_t)(n0 + nl) * CIN + tx] = tile[tx][nl];  // coalesced over c
    }
}

// ---------------------------------------------------------------------------
// Main kernel.  One wave32 owns 16 points and all 32 output channels:
//   D[32x16] = W[32x288] * G[288x16]  as 2 m-tiles x 72 k-steps of
//   V_WMMA_F32_16X16X4_F32 (2 x v8f accumulators).
// Neighbors are processed 3 at a time: one clause of 24 global_load_b64
// gathers, then 48 WMMAs with LDS weight reads interleaved.
//
// XT=true : gather from xt[b][n][c] (one b64 per lane per k-step).
// XT=false: fallback gather from x[b][c][n] (two strided b32 per lane).
//
// B-fragment striping mirrors the documented 32-bit A layout:
//   VGPR0: lanes 0-15 = K+0, lanes 16-31 = K+2 ; VGPR1: K+1 / K+3.
// ---------------------------------------------------------------------------
template<bool XT>
__global__ __launch_bounds__(256, 2)
void pointconv_wmma(const float* __restrict__ xsrc,
                    const int*   __restrict__ idx,
                    const float* __restrict__ weight,
                    const float* __restrict__ bias,
                    float*       __restrict__ out) {
    __shared__ float wlds[COUT * WSTR];

    const int tid  = threadIdx.x;
    const int lane = tid & 31;
    const int wave = tid >> 5;
    const int b    = blockIdx.y;

    // Stage W[o][K=j*32+c] = weight[o][c][j] into LDS.  9216 = 36*256 exactly,
    // so no divergence (EXEC stays all-ones for the WMMAs below).
#pragma unroll 1
    for (int e = tid; e < COUT * KFLAT; e += 256) {
        int o  = e / KFLAT;
        int kf = e - o * KFLAT;
        int j  = kf >> 5;
        int c  = kf & 31;
        wlds[o * WSTR + kf] = weight[(o * CIN + c) * KNB + j];
    }
    __syncthreads();

    const int p  = lane & 15;   // point within wave tile / M row within m-tile
    const int hi = lane >> 4;   // half-wave select (K+2/K+3 rows)
    const int n0 = (blockIdx.x * 8 + wave) * 16;

    // Per-lane neighbor indices for point p (replicated across half-waves).
    int idxr[KNB];
    const int* ip = idx + ((size_t)b * NPTS + (n0 + p)) * KNB;
#pragma unroll
    for (int j = 0; j < KNB; ++j) idxr[j] = ip[j];

    v8f acc0 = {};  // couts 0..15
    v8f acc1 = {};  // couts 16..31

    const float* xb = xsrc + (size_t)b * (size_t)CIN * NPTS;

    // Process neighbors 3 at a time: batch 24 b64 gathers, then 48 WMMAs.
#pragma unroll 1
    for (int jo = 0; jo < KNB; jo += 3) {
        v2f bf[3][8];
#pragma unroll
        for (int jj = 0; jj < 3; ++jj) {
            const int j = jo + jj;
            if (XT) {
                const float* gp = xb + (size_t)idxr[j] * CIN + hi * 2;
#pragma unroll
                for (int q = 0; q < 8; ++q)
                    bf[jj][q] = *(const v2f*)(gp + q * 4);   // 8B aligned, imm offset
            } else {
                const int nI = idxr[j];
#pragma unroll
                for (int q = 0; q < 8; ++q) {
                    const int c = q * 4 + hi * 2;
                    bf[jj][q].x = xb[(size_t)c * NPTS + nI];
                    bf[jj][q].y = xb[(size_t)(c + 1) * NPTS + nI];
                }
            }
        }
#pragma unroll
        for (int jj = 0; jj < 3; ++jj) {
#pragma unroll
            for (int q = 0; q < 8; ++q) {
                const int ks = (jo + jj) * 32 + q * 4 + hi * 2;
                v2f af0 = *(const v2f*)(&wlds[p * WSTR + ks]);          // o = p
                v2f af1 = *(const v2f*)(&wlds[(p + 16) * WSTR + ks]);   // o = p+16
                acc0 = __builtin_amdgcn_wmma_f32_16x16x4_f32(
                           false, af0, false, bf[jj][q], (short)0, acc0, false, false);
                acc1 = __builtin_amdgcn_wmma_f32_16x16x4_f32(
                           false, af1, false, bf[jj][q], (short)0, acc1, false, false);
            }
        }
    }

    // D layout: VGPR v, lane l -> o = v + 8*hi (+16 for acc1), n = n0 + p.
    float* ob = out + (size_t)b * COUT * NPTS;
#pragma unroll
    for (int v = 0; v < 8; ++v) {
        const int o0 = v + hi * 8;
        const int o1 = o0 + 16;
        ob[(size_t)o0 * NPTS + n0 + p] = acc0[v] + bias[o0];
        ob[(size_t)o1 * NPTS + n0 + p] = acc1[v] + bias[o1];
    }
}

// ---------------------------------------------------------------------------
extern "C" void kernel_launch(void* const* d_in, const int* in_sizes, int n_in,
                              void* d_out, int out_size, void* d_ws, size_t ws_size,
                              hipStream_t stream) {
    const float* x    = (const float*)d_in[0];
    const int*   idx  = (const int*)  d_in[1];
    const float* w    = (const float*)d_in[2];
    const float* bias = (const float*)d_in[3];
    float*       out  = (float*)d_out;

    const size_t xt_bytes = (size_t)BATCH * NPTS * CIN * sizeof(float); // 67 MB

    dim3 grid(NPTS / 128, BATCH);   // 8 waves * 16 pts = 128 pts / block
    if (ws_size >= xt_bytes) {
        float* xt = (float*)d_ws;
        dim3 tb(32, 8);
        dim3 tg(NPTS / CIN, BATCH);
        transpose_xc<<<tg, tb, 0, stream>>>(x, xt);
        pointconv_wmma<true><<<grid, 256, 0, stream>>>(xt, idx, w, bias, out);
    } else {
        pointconv_wmma<false><<<grid, 256, 0, stream>>>(x, idx, w, bias, out);
    }
}